// TopKClassifer_79843442033167
// MI455X (gfx1250) — compile-verified
//
#include <hip/hip_runtime.h>
#include <math.h>

// ---------------------------------------------------------------------------
// Static problem sizes (from reference)
// ---------------------------------------------------------------------------
#define N0   65536
#define CIN  64
#define HDIM 128
#define E0   524288
#define M1   32768   // after pool 0
#define M2   16384   // after pool 1
#define M3   8192    // after pool 2
#define K1   6       // knn k for layer 1 graph
#define K2   8       // knn k for layer 2 graph

typedef float v2f __attribute__((ext_vector_type(2)));
typedef float v8f __attribute__((ext_vector_type(8)));

// ---------------------------------------------------------------------------
// Layer-0 aggregation: agg[dst] += x[src] over explicit edge list (atomics)
// ---------------------------------------------------------------------------
__global__ void edge_scatter(const float* __restrict__ x,
                             const int* __restrict__ ei,
                             float* __restrict__ agg)
{
    int t  = blockIdx.x * blockDim.x + threadIdx.x;   // E0*16 threads
    int e  = t >> 4;
    int c4 = (t & 15) * 4;
    if (e >= E0) return;
    int src = ei[e];
    int dst = ei[E0 + e];
    float4 v = *(const float4*)(x + (size_t)src * CIN + c4);
    atomicAdd(&agg[(size_t)dst * CIN + c4 + 0], v.x);
    atomicAdd(&agg[(size_t)dst * CIN + c4 + 1], v.y);
    atomicAdd(&agg[(size_t)dst * CIN + c4 + 2], v.z);
    atomicAdd(&agg[(size_t)dst * CIN + c4 + 3], v.w);
}

// ---------------------------------------------------------------------------
// Layer-1/2 aggregation: agg[q] = sum_j x[nbr[q][j]]  (pure gather, no atomics)
// ---------------------------------------------------------------------------
template <int KNN>
__global__ void knn_agg(const float* __restrict__ x,
                        const int* __restrict__ nbr,
                        float* __restrict__ agg, int M)
{
    int t  = blockIdx.x * blockDim.x + threadIdx.x;   // M*32 threads
    int q  = t >> 5;
    int c4 = (t & 31) * 4;
    if (q >= M) return;
    float4 s = make_float4(0.f, 0.f, 0.f, 0.f);
#pragma unroll
    for (int j = 0; j < KNN; ++j) {
        int nb = nbr[q * KNN + j];
        float4 v = *(const float4*)(x + (size_t)nb * HDIM + c4);
        s.x += v.x; s.y += v.y; s.z += v.z; s.w += v.w;
    }
    *(float4*)(agg + (size_t)q * HDIM + c4) = s;
}

// ---------------------------------------------------------------------------
// Fused dual GEMM + bias + ReLU using V_WMMA_F32_16X16X4_F32 (wave32).
//   out = relu([A1 | A2] @ [W1 ; W2] + bias),   W1:[C1,128], W2:[C2,128]
// Block = 256 threads = 8 waves; each wave computes a 16x128 output tile.
//
// Weights staged in LDS as k-PAIRS: pair-row p holds float2{W[2p][n],W[2p+1][n]}
// so each lane fetches its whole B fragment with ONE ds_load_b64 that lands in
// an aligned VGPR pair (direct WMMA operand, no repack movs). The A fragment
// is one 8-byte global load (k-pairs start even and never straddle A1/A2
// since C1 is a multiple of 64).
//
// ISA lane layout (16x16x4 f32):
//   A (16x4):  lane L: m=L%16, vgpr0=A[m][2*(L/16)],   vgpr1=A[m][2*(L/16)+1]
//   B (4x16):  lane L: n=L%16, vgpr0=B[2*(L/16)][n],   vgpr1=B[2*(L/16)+1][n]
//   C/D:       lane L, vgpr r: (m = r + 8*(L/16), n = L%16)
// ---------------------------------------------------------------------------
template <int C1, int C2>
__global__ void gemm_dual_bias_relu(const float* __restrict__ A1,
                                    const float* __restrict__ A2,
                                    const float* __restrict__ W1,
                                    const float* __restrict__ W2,
                                    const float* __restrict__ bias,
                                    float* __restrict__ out)
{
    constexpr int K  = C1 + C2;
    constexpr int KC = 64;                 // k-chunk
    static_assert(K % KC == 0, "K must be multiple of chunk");
    static_assert((C1 % KC) == 0, "pairs must not straddle A1/A2");
    __shared__ float Bs[KC * HDIM];        // 32 KB, pair-interleaved
    __shared__ float bs[HDIM];

    const int tid  = threadIdx.x;
    const int wv   = tid >> 5;
    const int lane = tid & 31;
    const int hl   = lane >> 4;            // half-wave: k-pair 0 vs 1 of quad
    const int l16  = lane & 15;
    const int rowBase = blockIdx.x * 128 + wv * 16;
    const int m = rowBase + l16;

    if (tid < HDIM) bs[tid] = bias[tid];

    v8f acc[8];
#pragma unroll
    for (int t = 0; t < 8; ++t)
#pragma unroll
        for (int r = 0; r < 8; ++r) acc[t][r] = 0.f;

    for (int kc = 0; kc < K; kc += KC) {
        __syncthreads();
        // stage chunk as k-pairs: ((float2*)Bs)[p*HDIM + n] = {W[kc+2p][n], W[kc+2p+1][n]}
        for (int i = tid; i < (KC / 2) * HDIM; i += 256) {
            int p  = i >> 7;               // pair index in chunk
            int hh = i & 127;
            int kk = kc + 2 * p;
            float w0 = (kk < C1) ? W1[kk * HDIM + hh]
                                 : W2[(kk - C1) * HDIM + hh];
            float w1 = (kk + 1 < C1) ? W1[(kk + 1) * HDIM + hh]
                                     : W2[(kk + 1 - C1) * HDIM + hh];
            *(float2*)(Bs + (size_t)p * (2 * HDIM) + 2 * hh) = make_float2(w0, w1);
        }
        __syncthreads();

        for (int k0 = 0; k0 < KC; k0 += 4) {
            const int kA0 = kc + k0 + 2 * hl;        // even, pair-aligned
            const float* ap = (kA0 < C1)
                ? (A1 + (size_t)m * C1 + kA0)
                : (A2 + (size_t)m * C2 + (kA0 - C1));
            v2f a = *(const v2f*)ap;                 // one global_load_b64

            const int prow = (k0 >> 1) + hl;         // pair-row in LDS
#pragma unroll
            for (int t = 0; t < 8; ++t) {
                v2f b = *(const v2f*)(Bs + (size_t)prow * (2 * HDIM)
                                         + 2 * (t * 16 + l16));  // ds_load_b64
                acc[t] = __builtin_amdgcn_wmma_f32_16x16x4_f32(
                    false, a, false, b, (short)0, acc[t], false, false);
            }
        }
    }

#pragma unroll
    for (int t = 0; t < 8; ++t)
#pragma unroll
        for (int r = 0; r < 8; ++r) {
            int mm = rowBase + r + 8 * hl;
            int n  = t * 16 + l16;
            float v = acc[t][r] + bs[n];
            out[(size_t)mm * HDIM + n] = v > 0.f ? v : 0.f;
        }
}

// ---------------------------------------------------------------------------
// score[n] = tanh(dot(y[n], pw) / ||pw||); also emit order-preserving uint key
// ---------------------------------------------------------------------------
__global__ void score_kernel(const float* __restrict__ y,
                             const float* __restrict__ pw,
                             float* __restrict__ score,
                             unsigned* __restrict__ key, int n)
{
    int wv = threadIdx.x >> 5, lane = threadIdx.x & 31;
    float4 w4 = ((const float4*)pw)[lane];
    float nn = w4.x * w4.x + w4.y * w4.y + w4.z * w4.z + w4.w * w4.w;
    for (int o = 16; o; o >>= 1) nn += __shfl_down(nn, o, 32);
    float norm = sqrtf(__shfl(nn, 0, 32));
    int nwaves = gridDim.x * (blockDim.x >> 5);
    for (int row = blockIdx.x * (blockDim.x >> 5) + wv; row < n; row += nwaves) {
        float4 v = ((const float4*)(y + (size_t)row * HDIM))[lane];
        float d = v.x * w4.x + v.y * w4.y + v.z * w4.z + v.w * w4.w;
        for (int o = 16; o; o >>= 1) d += __shfl_down(d, o, 32);
        if (lane == 0) {
            float s = tanhf(d / norm);
            score[row] = s;
            unsigned u = __float_as_uint(s);
            key[row] = (u & 0x80000000u) ? ~u : (u | 0x80000000u);
        }
    }
}

// ---------------------------------------------------------------------------
// Exact top-k threshold via 4-pass MSB radix select (k = n/2)
// ---------------------------------------------------------------------------
struct RadixState {
    unsigned prefix, remaining, thrKey, tiesNeeded;
    unsigned hist[256];
};

__global__ void radix_init(RadixState* st, unsigned k)
{
    int t = threadIdx.x;
    if (t == 0) { st->prefix = 0; st->remaining = k; st->thrKey = 0; st->tiesNeeded = 0; }
    st->hist[t] = 0;
}

__global__ void radix_hist(const unsigned* __restrict__ key, int n,
                           RadixState* st, int pass)
{
    int shift = 24 - 8 * pass;
    unsigned mask = pass ? (0xFFFFFFFFu << (32 - 8 * pass)) : 0u;
    unsigned pref = st->prefix & mask;
    for (int i = blockIdx.x * blockDim.x + threadIdx.x; i < n;
         i += gridDim.x * blockDim.x) {
        unsigned kk = key[i];
        if ((kk & mask) == pref)
            atomicAdd(&st->hist[(kk >> shift) & 0xFFu], 1u);
    }
}

__global__ void radix_select(RadixState* st, int pass)
{
    __shared__ unsigned h[256];
    int t = threadIdx.x;
    h[t] = st->hist[t];
    __syncthreads();
    if (t == 0) {
        unsigned rem = st->remaining, cum = 0;
        int b = 255;
        for (; b > 0; --b) {
            if (cum + h[b] >= rem) break;
            cum += h[b];
        }
        st->prefix |= ((unsigned)b) << (24 - 8 * pass);
        st->remaining = rem - cum;
        if (pass == 3) { st->thrKey = st->prefix; st->tiesNeeded = rem - cum; }
    }
    __syncthreads();
    st->hist[t] = 0;   // ready for next pass
}

// ---------------------------------------------------------------------------
// Stable compaction of kept nodes (node order), deterministic tie quota
// ---------------------------------------------------------------------------
__global__ void compact_count(const unsigned* __restrict__ key,
                              const RadixState* st,
                              unsigned* __restrict__ cntG,
                              unsigned* __restrict__ cntE)
{
    __shared__ unsigned sg[8], se[8];
    unsigned thr = st->thrKey;
    int i = blockIdx.x * 256 + threadIdx.x;
    unsigned kk = key[i];
    unsigned long long mg = __ballot(kk > thr);
    unsigned long long me = __ballot(kk == thr);
    int wv = threadIdx.x >> 5, ln = threadIdx.x & 31;
    if (ln == 0) { sg[wv] = (unsigned)__popcll(mg); se[wv] = (unsigned)__popcll(me); }
    __syncthreads();
    if (threadIdx.x == 0) {
        unsigned g = 0, e = 0;
        for (int w = 0; w < 8; ++w) { g += sg[w]; e += se[w]; }
        cntG[blockIdx.x] = g; cntE[blockIdx.x] = e;
    }
}

__global__ void compact_scan(const unsigned* __restrict__ cntG,
                             const unsigned* __restrict__ cntE, int nBlocks,
                             const RadixState* st,
                             unsigned* __restrict__ eqOff,
                             unsigned* __restrict__ keptOff)
{
    if (threadIdx.x == 0) {
        unsigned ties = st->tiesNeeded;
        unsigned eAcc = 0, kAcc = 0;
        for (int b = 0; b < nBlocks; ++b) {
            eqOff[b] = eAcc; keptOff[b] = kAcc;
            unsigned e = cntE[b];
            unsigned take = (eAcc >= ties) ? 0u
                          : ((e < ties - eAcc) ? e : (ties - eAcc));
            kAcc += cntG[b] + take;
            eAcc += e;
        }
    }
}

__global__ void compact_write(const unsigned* __restrict__ key,
                              const RadixState* st,
                              const unsigned* __restrict__ eqOff,
                              const unsigned* __restrict__ keptOff,
                              const float2* __restrict__ posIn,
                              int* __restrict__ idxMap,
                              float2* __restrict__ posOut)
{
    __shared__ unsigned se[8], sk[8];
    unsigned thr = st->thrKey, ties = st->tiesNeeded;
    int i = blockIdx.x * 256 + threadIdx.x;
    unsigned kk = key[i];
    bool g = kk > thr, e = (kk == thr);
    int wv = threadIdx.x >> 5, ln = threadIdx.x & 31;
    unsigned long long lm = (1ull << ln) - 1ull;

    unsigned long long me = __ballot(e);
    if (ln == 0) se[wv] = (unsigned)__popcll(me);
    __syncthreads();
    unsigned eqBefore = eqOff[blockIdx.x] + (unsigned)__popcll(me & lm);
    for (int w = 0; w < wv; ++w) eqBefore += se[w];

    bool keep = g || (e && eqBefore < ties);
    unsigned long long mk = __ballot(keep);
    if (ln == 0) sk[wv] = (unsigned)__popcll(mk);
    __syncthreads();
    unsigned rank = keptOff[blockIdx.x] + (unsigned)__popcll(mk & lm);
    for (int w = 0; w < wv; ++w) rank += sk[w];

    if (keep) { idxMap[rank] = i; posOut[rank] = posIn[i]; }
}

__global__ void gather_gate(const float* __restrict__ y,
                            const float* __restrict__ score,
                            const int* __restrict__ idxMap,
                            float* __restrict__ xOut, int Mout)
{
    int t  = blockIdx.x * blockDim.x + threadIdx.x;  // Mout*32 threads
    int j  = t >> 5;
    int c4 = (t & 31) * 4;
    if (j >= Mout) return;
    int n = idxMap[j];
    float s = score[n];
    float4 v = *(const float4*)(y + (size_t)n * HDIM + c4);
    v.x *= s; v.y *= s; v.z *= s; v.w *= s;
    *(float4*)(xOut + (size_t)j * HDIM + c4) = v;
}

// ---------------------------------------------------------------------------
// Global max/mean pool (per channel), accumulated into h[256] across layers
// ---------------------------------------------------------------------------
__global__ void pool_partial(const float* __restrict__ x, int rowsPerBlock,
                             float* __restrict__ pmax, float* __restrict__ psum)
{
    int c  = threadIdx.x;                 // 128 threads
    int r0 = blockIdx.x * rowsPerBlock;
    float mx = -INFINITY, sm = 0.f;
    for (int r = 0; r < rowsPerBlock; ++r) {
        float v = x[(size_t)(r0 + r) * HDIM + c];
        mx = fmaxf(mx, v); sm += v;
    }
    pmax[blockIdx.x * HDIM + c] = mx;
    psum[blockIdx.x * HDIM + c] = sm;
}

__global__ void pool_finish(const float* __restrict__ pmax,
                            const float* __restrict__ psum, int nb,
                            float invM, float* __restrict__ h)
{
    int c = threadIdx.x;                  // 128 threads
    float mx = -INFINITY, sm = 0.f;
    for (int b = 0; b < nb; ++b) {
        mx = fmaxf(mx, pmax[b * HDIM + c]);
        sm += psum[b * HDIM + c];
    }
    h[c]        += mx;
    h[HDIM + c] += sm * invM;
}

// ---------------------------------------------------------------------------
// Brute-force KNN on 2-D positions; register-resident k-best insertion
// ---------------------------------------------------------------------------
template <int KNN>
__global__ void knn_kernel(const float2* __restrict__ pos, int M,
                           int* __restrict__ nbr)
{
    constexpr int TILE = 1024;
    __shared__ float2 tp[TILE];           // 8 KB
    int q = blockIdx.x * blockDim.x + threadIdx.x;
    float2 myp = pos[q];
    float bd[KNN]; int bi[KNN];
#pragma unroll
    for (int j = 0; j < KNN; ++j) { bd[j] = INFINITY; bi[j] = -1; }

    for (int base = 0; base < M; base += TILE) {
        __syncthreads();
        for (int i = threadIdx.x; i < TILE; i += blockDim.x) tp[i] = pos[base + i];
        __syncthreads();
        for (int i = 0; i < TILE; ++i) {
            int idx = base + i;
            float dx = myp.x - tp[i].x;
            float dy = myp.y - tp[i].y;
            float d = dx * dx + dy * dy;
            if (idx != q && d < bd[KNN - 1]) {
                float cd = d; int ci = idx;
#pragma unroll
                for (int j = 0; j < KNN; ++j) {
                    if (cd < bd[j]) {
                        float td = bd[j]; int ti = bi[j];
                        bd[j] = cd; bi[j] = ci; cd = td; ci = ti;
                    }
                }
            }
        }
    }
#pragma unroll
    for (int j = 0; j < KNN; ++j) nbr[q * KNN + j] = bi[j];
}

// ---------------------------------------------------------------------------
// Final MLP on the single pooled row: 256 -> 128 -> 64 -> 10
// ---------------------------------------------------------------------------
__global__ void mlp_kernel(const float* __restrict__ h,
                           const float* __restrict__ W1, const float* __restrict__ b1,
                           const float* __restrict__ W2, const float* __restrict__ b2,
                           const float* __restrict__ W3, const float* __restrict__ b3,
                           float* __restrict__ out)
{
    __shared__ float hh[256], t1[128], t2[64];
    int t = threadIdx.x;                  // 256 threads
    hh[t] = h[t];
    __syncthreads();
    if (t < 128) {
        float a = b1[t];
        for (int i = 0; i < 256; ++i) a += hh[i] * W1[i * 128 + t];
        t1[t] = fmaxf(a, 0.f);
    }
    __syncthreads();
    if (t < 64) {
        float a = b2[t];
        for (int i = 0; i < 128; ++i) a += t1[i] * W2[i * 64 + t];
        t2[t] = fmaxf(a, 0.f);
    }
    __syncthreads();
    if (t < 10) {
        float a = b3[t];
        for (int i = 0; i < 64; ++i) a += t2[i] * W3[i * 10 + t];
        out[t] = a;
    }
}

// ---------------------------------------------------------------------------
// Workspace layout
// ---------------------------------------------------------------------------
static constexpr size_t al(size_t x) { return (x + 255) & ~(size_t)255; }
static constexpr size_t OFF_AGG0 = 0;
static constexpr size_t OFF_Y0   = al(OFF_AGG0 + (size_t)N0 * CIN * 4);
static constexpr size_t OFF_SC   = al(OFF_Y0   + (size_t)N0 * HDIM * 4);
static constexpr size_t OFF_KEY  = al(OFF_SC   + (size_t)N0 * 4);
static constexpr size_t OFF_X1   = al(OFF_KEY  + (size_t)N0 * 4);
static constexpr size_t OFF_P1   = al(OFF_X1   + (size_t)M1 * HDIM * 4);
static constexpr size_t OFF_NB1  = al(OFF_P1   + (size_t)M1 * 8);
static constexpr size_t OFF_AGG1 = al(OFF_NB1  + (size_t)M1 * K1 * 4);
static constexpr size_t OFF_Y1   = al(OFF_AGG1 + (size_t)M1 * HDIM * 4);
static constexpr size_t OFF_X2   = al(OFF_Y1   + (size_t)M1 * HDIM * 4);
static constexpr size_t OFF_P2   = al(OFF_X2   + (size_t)M2 * HDIM * 4);
static constexpr size_t OFF_NB2  = al(OFF_P2   + (size_t)M2 * 8);
static constexpr size_t OFF_AGG2 = al(OFF_NB2  + (size_t)M2 * K2 * 4);
static constexpr size_t OFF_Y2   = al(OFF_AGG2 + (size_t)M2 * HDIM * 4);
static constexpr size_t OFF_X3   = al(OFF_Y2   + (size_t)M2 * HDIM * 4);
static constexpr size_t OFF_P3   = al(OFF_X3   + (size_t)M3 * HDIM * 4);
static constexpr size_t OFF_H    = al(OFF_P3   + (size_t)M3 * 8);
static constexpr size_t OFF_PMAX = al(OFF_H    + 256 * 4);
static constexpr size_t OFF_PSUM = al(OFF_PMAX + 256 * HDIM * 4);
static constexpr size_t OFF_RAD  = al(OFF_PSUM + 256 * HDIM * 4);
static constexpr size_t OFF_CG   = al(OFF_RAD  + sizeof(RadixState));
static constexpr size_t OFF_CE   = al(OFF_CG   + 256 * 4);
static constexpr size_t OFF_EQO  = al(OFF_CE   + 256 * 4);
static constexpr size_t OFF_KPO  = al(OFF_EQO  + 256 * 4);
static constexpr size_t OFF_MAP  = al(OFF_KPO  + 256 * 4);
// OFF_MAP + M1*4 = total (~132 MB)

// ---------------------------------------------------------------------------
// Host-side helper: full TopK pool step (select + stable compact + gate)
// ---------------------------------------------------------------------------
static void topk_pool(const float* y, const float* pwv, const float2* posIn,
                      int n, int kKeep, char* ws,
                      float* xOut, float2* posOut, hipStream_t stream)
{
    float*      score = (float*)(ws + OFF_SC);
    unsigned*   key   = (unsigned*)(ws + OFF_KEY);
    RadixState* st    = (RadixState*)(ws + OFF_RAD);
    unsigned*   cntG  = (unsigned*)(ws + OFF_CG);
    unsigned*   cntE  = (unsigned*)(ws + OFF_CE);
    unsigned*   eqOff = (unsigned*)(ws + OFF_EQO);
    unsigned*   kpOff = (unsigned*)(ws + OFF_KPO);
    int*        map   = (int*)(ws + OFF_MAP);

    score_kernel<<<256, 256, 0, stream>>>(y, pwv, score, key, n);
    radix_init<<<1, 256, 0, stream>>>(st, (unsigned)kKeep);
    for (int p = 0; p < 4; ++p) {
        radix_hist<<<256, 256, 0, stream>>>(key, n, st, p);
        radix_select<<<1, 256, 0, stream>>>(st, p);
    }
    int nb = n / 256;
    compact_count<<<nb, 256, 0, stream>>>(key, st, cntG, cntE);
    compact_scan<<<1, 1, 0, stream>>>(cntG, cntE, nb, st, eqOff, kpOff);
    compact_write<<<nb, 256, 0, stream>>>(key, st, eqOff, kpOff, posIn, map, posOut);
    gather_gate<<<(kKeep * 32) / 256, 256, 0, stream>>>(y, score, map, xOut, kKeep);
}

static void pool_stats(const float* x, int M, char* ws, hipStream_t stream)
{
    float* pmax = (float*)(ws + OFF_PMAX);
    float* psum = (float*)(ws + OFF_PSUM);
    float* h    = (float*)(ws + OFF_H);
    pool_partial<<<256, 128, 0, stream>>>(x, M / 256, pmax, psum);
    pool_finish<<<1, 128, 0, stream>>>(pmax, psum, 256, 1.f / (float)M, h);
}

// ---------------------------------------------------------------------------
extern "C" void kernel_launch(void* const* d_in, const int* in_sizes, int n_in,
                              void* d_out, int out_size, void* d_ws, size_t ws_size,
                              hipStream_t stream)
{
    (void)in_sizes; (void)n_in; (void)out_size; (void)ws_size;

    const float*  x     = (const float*)d_in[0];
    const float2* pos   = (const float2*)d_in[1];
    const int*    ei    = (const int*)d_in[2];
    const float*  Wrel0 = (const float*)d_in[4];
    const float*  brel0 = (const float*)d_in[5];
    const float*  Wroot0= (const float*)d_in[6];
    const float*  pw0   = (const float*)d_in[7];
    const float*  Wrel1 = (const float*)d_in[8];
    const float*  brel1 = (const float*)d_in[9];
    const float*  Wroot1= (const float*)d_in[10];
    const float*  pw1   = (const float*)d_in[11];
    const float*  Wrel2 = (const float*)d_in[12];
    const float*  brel2 = (const float*)d_in[13];
    const float*  Wroot2= (const float*)d_in[14];
    const float*  pw2   = (const float*)d_in[15];
    const float*  l1w   = (const float*)d_in[16];
    const float*  l1b   = (const float*)d_in[17];
    const float*  l2w   = (const float*)d_in[18];
    const float*  l2b   = (const float*)d_in[19];
    const float*  l3w   = (const float*)d_in[20];
    const float*  l3b   = (const float*)d_in[21];

    char* ws = (char*)d_ws;
    float*  agg0 = (float*)(ws + OFF_AGG0);
    float*  y0   = (float*)(ws + OFF_Y0);
    float*  x1   = (float*)(ws + OFF_X1);
    float2* p1   = (float2*)(ws + OFF_P1);
    int*    nb1  = (int*)(ws + OFF_NB1);
    float*  agg1 = (float*)(ws + OFF_AGG1);
    float*  y1   = (float*)(ws + OFF_Y1);
    float*  x2   = (float*)(ws + OFF_X2);
    float2* p2   = (float2*)(ws + OFF_P2);
    int*    nb2  = (int*)(ws + OFF_NB2);
    float*  agg2 = (float*)(ws + OFF_AGG2);
    float*  y2   = (float*)(ws + OFF_Y2);
    float*  x3   = (float*)(ws + OFF_X3);
    float2* p3   = (float2*)(ws + OFF_P3);
    float*  h    = (float*)(ws + OFF_H);

    hipMemsetAsync(agg0, 0, (size_t)N0 * CIN * 4, stream);
    hipMemsetAsync(h,    0, 256 * 4, stream);

    // ---- Layer 0 ----
    edge_scatter<<<(E0 * 16) / 256, 256, 0, stream>>>(x, ei, agg0);
    gemm_dual_bias_relu<CIN, CIN><<<N0 / 128, 256, 0, stream>>>(
        agg0, x, Wrel0, Wroot0, brel0, y0);
    topk_pool(y0, pw0, pos, N0, M1, ws, x1, p1, stream);
    pool_stats(x1, M1, ws, stream);
    knn_kernel<K1><<<M1 / 256, 256, 0, stream>>>(p1, M1, nb1);

    // ---- Layer 1 ----
    knn_agg<K1><<<(M1 * 32) / 256, 256, 0, stream>>>(x1, nb1, agg1, M1);
    gemm_dual_bias_relu<HDIM, HDIM><<<M1 / 128, 256, 0, stream>>>(
        agg1, x1, Wrel1, Wroot1, brel1, y1);
    topk_pool(y1, pw1, p1, M1, M2, ws, x2, p2, stream);
    pool_stats(x2, M2, ws, stream);
    knn_kernel<K2><<<M2 / 256, 256, 0, stream>>>(p2, M2, nb2);

    // ---- Layer 2 ----
    knn_agg<K2><<<(M2 * 32) / 256, 256, 0, stream>>>(x2, nb2, agg2, M2);
    gemm_dual_bias_relu<HDIM, HDIM><<<M2 / 128, 256, 0, stream>>>(
        agg2, x2, Wrel2, Wroot2, brel2, y2);
    topk_pool(y2, pw2, p2, M2, M3, ws, x3, p3, stream);
    pool_stats(x3, M3, ws, stream);

    // ---- Head ----
    mlp_kernel<<<1, 256, 0, stream>>>(h, l1w, l1b, l2w, l2b, l3w, l3b,
                                      (float*)d_out);
}